// EdgeNetEMD_8177617731795
// MI455X (gfx1250) — compile-verified
//
#include <hip/hip_runtime.h>

// ---------------------------------------------------------------------------
// EdgeConv GNN autoencoder for MI455X (gfx1250), f32 WMMA 16x16x4 formulation.
// M = 16 edges per wave-tile, N = output features, K = input features (x4).
// ---------------------------------------------------------------------------

typedef float v2f __attribute__((ext_vector_type(2)));
typedef float v8f __attribute__((ext_vector_type(8)));

#define BN_EPS 1e-5f

__device__ __forceinline__ v8f wmma4(v2f a, v2f b, v8f c) {
  // D = A(16x4 f32) * B(4x16 f32) + C(16x16 f32)
  return __builtin_amdgcn_wmma_f32_16x16x4_f32(false, a, false, b, (short)0, c,
                                               false, false);
}

__device__ __forceinline__ void wave_sync() {
  __builtin_amdgcn_wave_barrier();
}

// D-format (edge in VGPR idx, feat in lane) -> A-format (edge in lane,
// K-chunk in VGPR) via per-wave LDS tile, stride 36 floats (bank-conflict-free).
__device__ __forceinline__ void d_to_a_transpose(float* T, int lm, int half,
                                                 const v8f& C0, const v8f& C1,
                                                 v2f (&A)[8]) {
  wave_sync();
#pragma unroll
  for (int v = 0; v < 8; ++v) {
    T[(v + 8 * half) * 36 + lm]      = C0[v];
    T[(v + 8 * half) * 36 + 16 + lm] = C1[v];
  }
  wave_sync();
#pragma unroll
  for (int c = 0; c < 8; ++c) {
    const float2 t2 = *(const float2*)(T + lm * 36 + 4 * c + 2 * half);
    A[c].x = t2.x;
    A[c].y = t2.y;
  }
  wave_sync();
}

// ---------------------------------------------------------------------------
// EdgeConv: per edge e, in = [x[dst], x[src]-x[dst]] (2*IN_D),
// MLP (2*IN_D)->32 relu ->32 relu -> OUT_D (opt relu), atomicAdd into agg[dst].
// ---------------------------------------------------------------------------
template <int IN_D, int OUT_D, bool RELU_LAST, int WAVES>
__global__ void __launch_bounds__(WAVES * 32)
edgeconv_wmma(const float* __restrict__ xf, const int* __restrict__ src,
              const int* __restrict__ dst, const float* __restrict__ W1,
              const float* __restrict__ b1, const float* __restrict__ W2,
              const float* __restrict__ b2, const float* __restrict__ W3,
              const float* __restrict__ b3, float* __restrict__ agg, int E) {
  constexpr int NC1 = (2 * IN_D) / 4;  // layer-1 K chunks (enc: 2, dec: 1)

  const int lane = threadIdx.x & 31;
  const int wave = threadIdx.x >> 5;
  const int half = lane >> 4;   // 0: K pair {0,1}; 1: K pair {2,3}
  const int lm   = lane & 15;   // edge-in-tile (A) / feature (B,D) index

  __shared__ float Tsh[WAVES][16 * 36];
  float* T = Tsh[wave];

  // ---- loop-invariant weights in B-format registers -----------------------
  v2f B1r[NC1][2];
#pragma unroll
  for (int c = 0; c < NC1; ++c)
#pragma unroll
    for (int t = 0; t < 2; ++t) {
      const int k = 4 * c + 2 * half, n = 16 * t + lm;
      B1r[c][t].x = W1[k * 32 + n];
      B1r[c][t].y = W1[(k + 1) * 32 + n];
    }
  v2f B2r[8][2];
#pragma unroll
  for (int c = 0; c < 8; ++c)
#pragma unroll
    for (int t = 0; t < 2; ++t) {
      const int k = 4 * c + 2 * half, n = 16 * t + lm;
      B2r[c][t].x = W2[k * 32 + n];
      B2r[c][t].y = W2[(k + 1) * 32 + n];
    }
  v2f B3r[8];
#pragma unroll
  for (int c = 0; c < 8; ++c) {
    const int k = 4 * c + 2 * half;
    B3r[c].x = (lm < OUT_D) ? W3[k * OUT_D + lm] : 0.f;
    B3r[c].y = (lm < OUT_D) ? W3[(k + 1) * OUT_D + lm] : 0.f;
  }
  const float bias1[2] = {b1[lm], b1[16 + lm]};
  const float bias2[2] = {b2[lm], b2[16 + lm]};
  const float bias3    = (lm < OUT_D) ? b3[lm] : 0.f;

  const int ntiles = (E + 15) >> 4;
  for (int tile = blockIdx.x * WAVES + wave; tile < ntiles;
       tile += gridDim.x * WAVES) {
    const int ebase = tile << 4;
    int e = ebase + lm;
    if (e >= E) e = E - 1;  // clamp; tail lanes masked at the atomic stage
    const int di = dst[e];
    const int si = src[e];

    // ---- build layer-1 A tiles directly from gathered features -----------
    v2f A1[NC1];
    if constexpr (IN_D == 4) {
      const float2 xi = *(const float2*)(xf + (size_t)di * 4 + half * 2);
      const float2 xj = *(const float2*)(xf + (size_t)si * 4 + half * 2);
      A1[0].x = xi.x;        A1[0].y = xi.y;         // [x_i]
      A1[1].x = xj.x - xi.x; A1[1].y = xj.y - xi.y;  // [x_j - x_i]
    } else {  // IN_D == 2: single K-chunk, halves split xi vs (xj-xi)
      const float2 xi = *(const float2*)(xf + (size_t)di * 2);
      const float2 xj = *(const float2*)(xf + (size_t)si * 2);
      if (half) { A1[0].x = xj.x - xi.x; A1[0].y = xj.y - xi.y; }
      else      { A1[0].x = xi.x;        A1[0].y = xi.y; }
    }

    // ---- layer 1: (2*IN_D) -> 32, ReLU ------------------------------------
    v8f C0, C1;
#pragma unroll
    for (int v = 0; v < 8; ++v) { C0[v] = bias1[0]; C1[v] = bias1[1]; }
#pragma unroll
    for (int c = 0; c < NC1; ++c) {
      C0 = wmma4(A1[c], B1r[c][0], C0);
      C1 = wmma4(A1[c], B1r[c][1], C1);
    }
#pragma unroll
    for (int v = 0; v < 8; ++v) {
      C0[v] = fmaxf(C0[v], 0.f);
      C1[v] = fmaxf(C1[v], 0.f);
    }

    v2f A2[8];
    d_to_a_transpose(T, lm, half, C0, C1, A2);

    // ---- layer 2: 32 -> 32, ReLU ------------------------------------------
#pragma unroll
    for (int v = 0; v < 8; ++v) { C0[v] = bias2[0]; C1[v] = bias2[1]; }
#pragma unroll
    for (int c = 0; c < 8; ++c) {
      C0 = wmma4(A2[c], B2r[c][0], C0);
      C1 = wmma4(A2[c], B2r[c][1], C1);
    }
#pragma unroll
    for (int v = 0; v < 8; ++v) {
      C0[v] = fmaxf(C0[v], 0.f);
      C1[v] = fmaxf(C1[v], 0.f);
    }

    v2f A3[8];
    d_to_a_transpose(T, lm, half, C0, C1, A3);

    // ---- layer 3: 32 -> OUT_D (padded to one 16-wide N tile) --------------
    v8f F;
#pragma unroll
    for (int v = 0; v < 8; ++v) F[v] = bias3;
#pragma unroll
    for (int c = 0; c < 8; ++c) F = wmma4(A3[c], B3r[c], F);
    if constexpr (RELU_LAST) {
#pragma unroll
      for (int v = 0; v < 8; ++v) F[v] = fmaxf(F[v], 0.f);
    }

    // ---- segment-sum: D[edge=v+8*half][feat=lm], only feat<OUT_D real -----
    if (lm < OUT_D) {
#pragma unroll
      for (int v = 0; v < 8; ++v) {
        const int ee = ebase + v + 8 * half;
        if (ee < E) {
          atomicAdd(agg + (size_t)dst[ee] * OUT_D + lm, F[v]);
        }
      }
    }
  }
}

// ---------------------------------------------------------------------------
// Small helper kernels
// ---------------------------------------------------------------------------
__global__ void zero_kernel(float* __restrict__ p, int n) {
  for (int i = blockIdx.x * blockDim.x + threadIdx.x; i < n;
       i += gridDim.x * blockDim.x)
    p[i] = 0.f;
}

__global__ void bn_stats_kernel(const float* __restrict__ x,
                                float* __restrict__ stats, int N) {
  float acc[8] = {0.f, 0.f, 0.f, 0.f, 0.f, 0.f, 0.f, 0.f};
  for (int i = blockIdx.x * blockDim.x + threadIdx.x; i < N;
       i += gridDim.x * blockDim.x) {
    const float4 v = ((const float4*)x)[i];
    acc[0] += v.x; acc[1] += v.y; acc[2] += v.z; acc[3] += v.w;
    acc[4] += v.x * v.x; acc[5] += v.y * v.y;
    acc[6] += v.z * v.z; acc[7] += v.w * v.w;
  }
#pragma unroll
  for (int m = 16; m >= 1; m >>= 1)
#pragma unroll
    for (int j = 0; j < 8; ++j) acc[j] += __shfl_xor(acc[j], m, 32);

  __shared__ float sh[8];
  if (threadIdx.x < 8) sh[threadIdx.x] = 0.f;
  __syncthreads();
  if ((threadIdx.x & 31) == 0) {
#pragma unroll
    for (int j = 0; j < 8; ++j) atomicAdd(&sh[j], acc[j]);
  }
  __syncthreads();
  if (threadIdx.x < 8) atomicAdd(&stats[threadIdx.x], sh[threadIdx.x]);
}

__global__ void bn_apply_kernel(const float* __restrict__ x,
                                const float* __restrict__ stats,
                                const float* __restrict__ gamma,
                                const float* __restrict__ beta,
                                float* __restrict__ xn, int N) {
  const float invN = 1.f / (float)N;
  float mean[4], sc[4], bt[4];
#pragma unroll
  for (int f = 0; f < 4; ++f) {
    const float m   = stats[f] * invN;
    const float var = fmaxf(stats[4 + f] * invN - m * m, 0.f);
    mean[f] = m;
    sc[f]   = rsqrtf(var + BN_EPS) * gamma[f];
    bt[f]   = beta[f];
  }
  for (int i = blockIdx.x * blockDim.x + threadIdx.x; i < N;
       i += gridDim.x * blockDim.x) {
    const float4 v = ((const float4*)x)[i];
    float4 o;
    o.x = (v.x - mean[0]) * sc[0] + bt[0];
    o.y = (v.y - mean[1]) * sc[1] + bt[1];
    o.z = (v.z - mean[2]) * sc[2] + bt[2];
    o.w = (v.w - mean[3]) * sc[3] + bt[3];
    ((float4*)xn)[i] = o;
  }
}

__global__ void count_kernel(const int* __restrict__ dst,
                             float* __restrict__ cnt, int E) {
  for (int e = blockIdx.x * blockDim.x + threadIdx.x; e < E;
       e += gridDim.x * blockDim.x)
    atomicAdd(&cnt[dst[e]], 1.f);
}

template <int C>
__global__ void div_kernel(const float* __restrict__ agg,
                           const float* __restrict__ cnt,
                           float* __restrict__ out, int N) {
  for (int i = blockIdx.x * blockDim.x + threadIdx.x; i < N;
       i += gridDim.x * blockDim.x) {
    const float inv = 1.f / fmaxf(cnt[i], 1.f);
#pragma unroll
    for (int c = 0; c < C; ++c)
      out[(size_t)i * C + c] = agg[(size_t)i * C + c] * inv;
  }
}

// ---------------------------------------------------------------------------
extern "C" void kernel_launch(void* const* d_in, const int* in_sizes, int n_in,
                              void* d_out, int out_size, void* d_ws,
                              size_t ws_size, hipStream_t stream) {
  const float* x     = (const float*)d_in[0];
  const int*   src   = (const int*)d_in[1];
  const int*   dst   = (const int*)d_in[2];
  const float* gamma = (const float*)d_in[3];
  const float* beta  = (const float*)d_in[4];
  const float* eW1 = (const float*)d_in[5];  const float* eb1 = (const float*)d_in[6];
  const float* eW2 = (const float*)d_in[7];  const float* eb2 = (const float*)d_in[8];
  const float* eW3 = (const float*)d_in[9];  const float* eb3 = (const float*)d_in[10];
  const float* dW1 = (const float*)d_in[11]; const float* db1 = (const float*)d_in[12];
  const float* dW2 = (const float*)d_in[13]; const float* db2 = (const float*)d_in[14];
  const float* dW3 = (const float*)d_in[15]; const float* db3 = (const float*)d_in[16];

  const int N = in_sizes[0] / 4;
  const int E = in_sizes[1];

  // workspace layout (floats): [stats 8 | xn 4N | agg1 2N | h2 2N | agg2 4N | cnt N]
  float* ws    = (float*)d_ws;
  float* stats = ws;
  float* xn    = ws + 8;
  float* agg1  = xn + (size_t)4 * N;
  float* h2    = agg1 + (size_t)2 * N;
  float* agg2  = h2 + (size_t)2 * N;
  float* cnt   = agg2 + (size_t)4 * N;

  const int T = 256;
  zero_kernel<<<1, 32, 0, stream>>>(stats, 8);
  zero_kernel<<<1024, T, 0, stream>>>(agg1, 2 * N);
  zero_kernel<<<1024, T, 0, stream>>>(agg2, 4 * N);
  zero_kernel<<<1024, T, 0, stream>>>(cnt, N);

  bn_stats_kernel<<<64, T, 0, stream>>>(x, stats, N);
  bn_apply_kernel<<<512, T, 0, stream>>>(x, stats, gamma, beta, xn, N);
  count_kernel<<<2048, T, 0, stream>>>(dst, cnt, E);

  constexpr int WAVES = 4;
  const int ntiles  = (E + 15) / 16;
  int eblocks = (ntiles + WAVES - 1) / WAVES;
  if (eblocks > 4096) eblocks = 4096;
  if (eblocks < 1) eblocks = 1;

  edgeconv_wmma<4, 2, true, WAVES><<<eblocks, WAVES * 32, 0, stream>>>(
      xn, src, dst, eW1, eb1, eW2, eb2, eW3, eb3, agg1, E);
  div_kernel<2><<<512, T, 0, stream>>>(agg1, cnt, h2, N);
  edgeconv_wmma<2, 4, false, WAVES><<<eblocks, WAVES * 32, 0, stream>>>(
      h2, src, dst, dW1, db1, dW2, db2, dW3, db3, agg2, E);
  div_kernel<4><<<512, T, 0, stream>>>(agg2, cnt, (float*)d_out, N);
}